// CompanionSSMBase_58188216926762
// MI455X (gfx1250) — compile-verified
//
#include <hip/hip_runtime.h>

// CompanionSSM on MI455X (gfx1250, wave32).
//   y[b,v,t] = sum_{s<=t} u[b,s,v] * k[v,t-s],  k = Krylov impulse response of
//   the 256x256 companion system. Conv implemented as a tiled lower-triangular
//   Toeplitz matmul on V_WMMA_F32_16X16X4_F32 with 2x4 register tiling.
//   Triangular masking is done by a 32-element zero pad in front of the LDS
//   kernel row -> the hot loop has zero compares/selects and one uniform trip.

typedef __attribute__((ext_vector_type(2))) float v2f;
typedef __attribute__((ext_vector_type(8))) float v8f;

#define N_VARS 64
#define LATENT 256
#define SEQ_L  2048
#define BATCH  128
#define KPAD   32   // leading zeros in LDS kernel row (covers lags down to -31)

// ---------------------------------------------------------------------------
// Kernel 1: transpose u(b,s,v) -> ut(v,s,b) so conv B-operand loads are
// contiguous in the batch dimension. One block per s; LDS tile 64x128 floats
// padded to stride 129 (129 % 64 == 1 -> conflict-free both phases).
// ---------------------------------------------------------------------------
__global__ __launch_bounds__(256) void ssm_transpose_u(const float* __restrict__ u,
                                                       float* __restrict__ ut) {
  const int s = blockIdx.x;  // 0..2047
  __shared__ float tile[N_VARS * 129];
  for (int i = threadIdx.x; i < N_VARS * BATCH; i += 256) {
    const int v = i & 63;           // consecutive tids -> consecutive v (coalesced read)
    const int b = i >> 6;
    tile[v * 129 + b] = u[((size_t)b * SEQ_L + s) * N_VARS + v];
  }
  __syncthreads();
  for (int i = threadIdx.x; i < N_VARS * BATCH; i += 256) {
    const int b = i & 127;          // consecutive tids -> consecutive b (coalesced write)
    const int v = i >> 7;
    ut[((size_t)v * SEQ_L + s) * BATCH + b] = tile[v * 129 + b];
  }
}

// ---------------------------------------------------------------------------
// Kernel 2: Krylov scan. One wave per variable v. State x (256) lives in 8
// VGPRs per lane: x[j], j = lane + 32*r. Companion update:
//   x'[j] = (j>=64 ? x[j-64] : 0) + coeff_n[j] * x[255]
// j-64 == lane + 32*(r-2): the shift is a pure register rename (free once the
// h-loop is unrolled by 4). Dot product reduced with 5 shfl_xor steps; the
// reduced value is wave-uniform, so lane 0 batches 4 of them into one b128
// store (one EXEC toggle per 4 steps).
// ---------------------------------------------------------------------------
__global__ __launch_bounds__(32) void ssm_krylov(const float* __restrict__ coeff,
                                                 const float* __restrict__ Bm,
                                                 const float* __restrict__ Cm,
                                                 float* __restrict__ kout) {
  const int v = blockIdx.x;
  const int lane = threadIdx.x;

  float cn[8], cv[8], x[8];

  // L1-normalize coeff (single row of 256): xn = sum|coeff|; mean_norm == xn.
  float s = 0.f;
#pragma unroll
  for (int r = 0; r < 8; ++r) { cn[r] = coeff[lane + 32 * r]; s += fabsf(cn[r]); }
#pragma unroll
  for (int m = 16; m >= 1; m >>= 1) s += __shfl_xor(s, m, 32);
  const float scale = (s > 1e-4f) ? (1.0f / (s + 1e-8f)) : 1.0f;
#pragma unroll
  for (int r = 0; r < 8; ++r) cn[r] *= scale;

#pragma unroll
  for (int r = 0; r < 8; ++r) x[r] = Bm[(lane + 32 * r) * N_VARS + v];   // b_v column
#pragma unroll
  for (int r = 0; r < 8; ++r) cv[r] = Cm[v * LATENT + lane + 32 * r];    // c_v row

  float* krow = kout + (size_t)v * SEQ_L;
  for (int h = 0; h < SEQ_L; h += 4) {
    float pq[4];
#pragma unroll
    for (int hh = 0; hh < 4; ++hh) {
      float p = 0.f;
#pragma unroll
      for (int r = 0; r < 8; ++r) p = fmaf(cv[r], x[r], p);
#pragma unroll
      for (int m = 16; m >= 1; m >>= 1) p += __shfl_xor(p, m, 32);
      pq[hh] = p;

      const float x255 = __shfl(x[7], 31, 32);   // broadcast old x[255]
#pragma unroll
      for (int r = 7; r >= 2; --r) x[r] = x[r - 2];  // shift by 64 = rename
      x[1] = 0.f; x[0] = 0.f;
#pragma unroll
      for (int r = 0; r < 8; ++r) x[r] = fmaf(cn[r], x255, x[r]);
    }
    if (lane == 0)
      *(float4*)(krow + h) = make_float4(pq[0], pq[1], pq[2], pq[3]);
  }
}

// ---------------------------------------------------------------------------
// Kernel 3: causal conv as blocked Toeplitz matmul on v_wmma_f32_16x16x4_f32.
// Grid (16 t-blocks, 64 v); block = 256 = 8 waves. Wave (wt,wb) owns a 2x4
// register tile: t-tiles {T0, T0+16} x batch-tiles {wb*64 + j*16}. A-operands
// (Toeplitz lags from LDS) are shared across the 4 batch tiles, B-operands
// (coalesced ut rows, immediate offsets) across the 2 t-tiles.
// Zero pad in kl2[0..31] makes every negative lag read 0 -> single uniform
// loop, no compares/selects, EXEC all-ones throughout (WMMA requirement).
// Operand layout per ISA (32-bit, wave32):
//   A 16x4 : lanes 0-15 -> {K0,K1}, lanes 16-31 -> {K2,K3}, M = lane&15
//   B 4x16 : mirrored, N = lane&15
//   C/D    : VGPR r -> M=r (lanes 0-15) / M=r+8 (lanes 16-31), N = lane&15
// ---------------------------------------------------------------------------
__global__ __launch_bounds__(256) void ssm_conv_wmma(const float* __restrict__ ut,
                                                     const float* __restrict__ kker,
                                                     float* __restrict__ out) {
  __shared__ float kl2[KPAD + SEQ_L];  // 8.125 KB: [zero pad | k row for v]
  const int tb    = (int)gridDim.x - 1 - (int)blockIdx.x;  // longest strips first
  const int Tbase = tb * 128;
  const int v     = blockIdx.y;

  // Stage: zero the pad (8 x b128) and bulk-copy the row as float4 (b128s).
  if (threadIdx.x < KPAD / 4)
    ((float4*)kl2)[threadIdx.x] = make_float4(0.f, 0.f, 0.f, 0.f);
  {
    const float4* src = (const float4*)(kker + (size_t)v * SEQ_L);
    float4* dst = (float4*)(kl2 + KPAD);
#pragma unroll
    for (int i = threadIdx.x; i < SEQ_L / 4; i += 256)
      dst[i] = src[i];
  }
  __syncthreads();

  const int lane = threadIdx.x & 31;
  const int wave = threadIdx.x >> 5;
  const int wt   = wave >> 1;        // t-pair within block (0..3)
  const int wb   = wave & 1;         // batch half (0..1)
  const int li   = lane & 15;
  const int hi16 = lane >> 4;        // half-wave: 0 or 1
  const int kq0  = hi16 * 2;         // K base for this half-wave

  const int T0   = Tbase + wt * 32;  // first t-tile; second is T0+16
  const int arow = T0 + li - kq0;    // folded A row index (lag base)
  const float* klp = kl2 + KPAD;     // klp[d] valid for d in [-KPAD, SEQ_L)

  v8f acc0[4] = {};                  // t-tile T0,    batch tiles j=0..3
  v8f acc1[4] = {};                  // t-tile T0+16, batch tiles j=0..3

  // B pointer: ut[v][kq0][wb*64 + li]; advance by 16 rows per s-tile.
  const float* pS = ut + ((size_t)v * SEQ_L + kq0) * BATCH + wb * 64 + li;

  // Single uniform loop covers both diagonals: lags below 0 hit the zero pad.
  for (int S0 = 0; S0 <= T0 + 16; S0 += 16, pS += 16 * BATCH) {
    __builtin_prefetch(pS + 32 * BATCH, 0, 1);  // global_prefetch_b8, 2 tiles ahead
#pragma unroll
    for (int kk = 0; kk < 4; ++kk) {
      const int d = arow - S0 - kk * 4;  // lag for q=kq0, in [-31, 2047]
      v2f a0, a1;
      a0.x = klp[d];      a0.y = klp[d - 1];   // adjacent -> ds_load_2addr
      a1.x = klp[d + 16]; a1.y = klp[d + 15];
      const float* up = pS + kk * 4 * BATCH;
#pragma unroll
      for (int j = 0; j < 4; ++j) {
        v2f b;
        b.x = up[j * 16];
        b.y = up[j * 16 + BATCH];
        acc0[j] = __builtin_amdgcn_wmma_f32_16x16x4_f32(false, a0, false, b,
                                                        (short)0, acc0[j], false, false);
        acc1[j] = __builtin_amdgcn_wmma_f32_16x16x4_f32(false, a1, false, b,
                                                        (short)0, acc1[j], false, false);
      }
    }
  }

  // ---- store: each lane owns 8 consecutive t's per tile -> b128 stores ----
#pragma unroll
  for (int j = 0; j < 4; ++j) {
    const int bb = wb * 64 + j * 16 + li;
    float* op0 = out + ((size_t)bb * N_VARS + v) * SEQ_L + T0 + hi16 * 8;
    *(float4*)(op0)     = make_float4(acc0[j][0], acc0[j][1], acc0[j][2], acc0[j][3]);
    *(float4*)(op0 + 4) = make_float4(acc0[j][4], acc0[j][5], acc0[j][6], acc0[j][7]);
    float* op1 = op0 + 16;
    *(float4*)(op1)     = make_float4(acc1[j][0], acc1[j][1], acc1[j][2], acc1[j][3]);
    *(float4*)(op1 + 4) = make_float4(acc1[j][4], acc1[j][5], acc1[j][6], acc1[j][7]);
  }
}

// ---------------------------------------------------------------------------
// Launch. Inputs (setup_inputs order): u (128*2048*64), coeff (256),
// B (256*64), C (64*256). Output: y (128*64*2048) f32.
// Workspace: k (512 KB) + ut (67 MB), both fully rewritten every call.
// ---------------------------------------------------------------------------
extern "C" void kernel_launch(void* const* d_in, const int* in_sizes, int n_in,
                              void* d_out, int out_size, void* d_ws, size_t ws_size,
                              hipStream_t stream) {
  (void)in_sizes; (void)n_in; (void)out_size; (void)ws_size;
  const float* u     = (const float*)d_in[0];
  const float* coeff = (const float*)d_in[1];
  const float* Bm    = (const float*)d_in[2];
  const float* Cm    = (const float*)d_in[3];
  float* out = (float*)d_out;

  float* kbuf = (float*)d_ws;                       // 64*2048 floats
  float* ut   = kbuf + (size_t)N_VARS * SEQ_L;      // 64*2048*128 floats

  ssm_transpose_u<<<SEQ_L, 256, 0, stream>>>(u, ut);
  ssm_krylov<<<N_VARS, 32, 0, stream>>>(coeff, Bm, Cm, kbuf);
  ssm_conv_wmma<<<dim3(SEQ_L / 128, N_VARS), 256, 0, stream>>>(ut, kbuf, out);
}